// S6Block_46368466927943
// MI455X (gfx1250) — compile-verified
//
#include <hip/hip_runtime.h>
#include <math.h>

// ---------------------------------------------------------------------------
// S6 / Mamba selective-scan block for MI455X (gfx1250), wave32.
//   BSZ=2, L=2048, DM=512, N=16, R=32
// Projections use V_WMMA_F32_16X16X4_F32 (full fp32 accuracy).
// Scan uses a 3-pass chunked linear-recurrence (16 chunks of 128).
// ---------------------------------------------------------------------------

typedef __attribute__((ext_vector_type(2))) float v2f;
typedef __attribute__((ext_vector_type(8))) float v8f;

#define BSZ   2
#define LSEQ  2048
#define DMOD  512
#define NST   16
#define RDT   32
#define MROWS (BSZ * LSEQ)        // 4096 rows for the projection GEMMs
#define CHUNK 128
#define NCH   (LSEQ / CHUNK)      // 16 chunks

// ---------------------------------------------------------------------------
// Kernel 1: fused projections.
//   BC  = X @ W_bc.T  + b_bc      -> Bm (.,0:16), Cm (.,16:32)
//   H   = X @ W_dt1.T + b_dt1     (kept in regs/LDS)
//   dt  = clip(softplus(H @ W_dt2.T + b_dt2))
// One wave per 16-row M tile; 4 waves / block; 64 blocks.
// WMMA f32 16x16x4 layouts:
//   A (16x4):  m = lane&15, holds K = {kb, kb+1}, kb = 2*(lane>>4)
//   B (4x16):  n = lane&15, holds K = {kb, kb+1}
//   C (16x16): vgpr v at lane L -> (M = v + 8*(L>>4), N = L&15)
// ---------------------------------------------------------------------------
__global__ __launch_bounds__(128) void s6_proj_kernel(
    const float* __restrict__ X,    const float* __restrict__ Wbc,
    const float* __restrict__ bbc,  const float* __restrict__ Wdt1,
    const float* __restrict__ bdt1, const float* __restrict__ Wdt2,
    const float* __restrict__ bdt2,
    float* __restrict__ Bm, float* __restrict__ Cm, float* __restrict__ dt)
{
    __shared__ float Hs[4][16 * 33];            // per-wave 16x32 H tile, padded

    const int lane = threadIdx.x & 31;
    const int wv   = threadIdx.x >> 5;
    const int tile = blockIdx.x * 4 + wv;       // 0..255
    const int row0 = tile * 16;
    const int jn   = lane & 15;                 // N / M-low index
    const int kb   = (lane >> 4) * 2;           // K sub-offset per half-wave
    const int mhi  = (lane >> 4) * 8;           // C-layout M offset

    const int   mrow = row0 + jn;
    const float* xrow  = X    + (size_t)mrow * DMOD;
    const float* wbc0  = Wbc  + (size_t)jn        * DMOD;  // W_bc rows 0..15 -> Bm
    const float* wbc1  = Wbc  + (size_t)(jn + 16) * DMOD;  // W_bc rows 16..31 -> Cm
    const float* wdtA  = Wdt1 + (size_t)jn        * DMOD;
    const float* wdtB  = Wdt1 + (size_t)(jn + 16) * DMOD;

    v8f accB = {}, accC = {}, accH0 = {}, accH1 = {};

    #pragma unroll 4
    for (int ks = 0; ks < DMOD / 4; ++ks) {
        const int k = ks * 4 + kb;
        v2f a  = *(const v2f*)(xrow + k);
        v2f b0 = *(const v2f*)(wbc0 + k);
        v2f b1 = *(const v2f*)(wbc1 + k);
        v2f b2 = *(const v2f*)(wdtA + k);
        v2f b3 = *(const v2f*)(wdtB + k);
        accB  = __builtin_amdgcn_wmma_f32_16x16x4_f32(false, a, false, b0, (short)0, accB,  false, false);
        accC  = __builtin_amdgcn_wmma_f32_16x16x4_f32(false, a, false, b1, (short)0, accC,  false, false);
        accH0 = __builtin_amdgcn_wmma_f32_16x16x4_f32(false, a, false, b2, (short)0, accH0, false, false);
        accH1 = __builtin_amdgcn_wmma_f32_16x16x4_f32(false, a, false, b3, (short)0, accH1, false, false);
    }

    const float biasB  = bbc[jn];
    const float biasC  = bbc[jn + 16];
    const float biasH0 = bdt1[jn];
    const float biasH1 = bdt1[jn + 16];

    #pragma unroll
    for (int v = 0; v < 8; ++v) {
        const int m = row0 + mhi + v;
        Bm[(size_t)m * NST + jn] = accB[v] + biasB;
        Cm[(size_t)m * NST + jn] = accC[v] + biasC;
        Hs[wv][(mhi + v) * 33 + jn]      = accH0[v] + biasH0;
        Hs[wv][(mhi + v) * 33 + jn + 16] = accH1[v] + biasH1;
    }
    // Same-wave LDS RAW: DS ops are in-order within a wave; compiler inserts
    // s_wait_dscnt for the returned data. No cross-wave sharing -> no barrier.

    // GEMM3: dt_tile(16 x 512) = H(16x32) @ W_dt2.T, then bias+softplus+clip.
    for (int pt = 0; pt < DMOD / 16; ++pt) {
        const int    prow = pt * 16 + jn;
        const float* w2   = Wdt2 + (size_t)prow * RDT;
        v8f acc = {};
        #pragma unroll
        for (int ks = 0; ks < RDT / 4; ++ks) {
            const int k = ks * 4 + kb;
            v2f a;
            a.x = Hs[wv][jn * 33 + k];
            a.y = Hs[wv][jn * 33 + k + 1];
            v2f b = *(const v2f*)(w2 + k);
            acc = __builtin_amdgcn_wmma_f32_16x16x4_f32(false, a, false, b, (short)0, acc, false, false);
        }
        const float bias2 = bdt2[prow];
        #pragma unroll
        for (int v = 0; v < 8; ++v) {
            const int m = row0 + mhi + v;
            float z  = acc[v] + bias2;
            float sp = (z > 20.f) ? z : __logf(1.f + __expf(z));   // softplus
            sp = fminf(fmaxf(sp, 1e-7f), 1e6f);
            dt[(size_t)m * DMOD + prow] = sp;
        }
    }
}

// ---------------------------------------------------------------------------
// Kernel 2 (pass A): per-chunk partial scan. Thread = (b, chunk, d), holding
// all 16 states in registers. Computes P = prod(a) and S = state-from-zero.
// ---------------------------------------------------------------------------
__global__ __launch_bounds__(256) void s6_scan_partial(
    const float* __restrict__ X,  const float* __restrict__ Ap,
    const float* __restrict__ Bm, const float* __restrict__ dt,
    float* __restrict__ Pc, float* __restrict__ Sc)
{
    const int tid = blockIdx.x * blockDim.x + threadIdx.x;   // BSZ*NCH*DMOD
    const int d = tid % DMOD;
    const int c = (tid / DMOD) % NCH;
    const int b = tid / (DMOD * NCH);

    float An[NST];
    #pragma unroll
    for (int n = 0; n < NST; ++n) An[n] = Ap[d * NST + n];

    float P[NST], S[NST];
    #pragma unroll
    for (int n = 0; n < NST; ++n) { P[n] = 1.f; S[n] = 0.f; }

    const size_t l0 = (size_t)b * LSEQ + (size_t)c * CHUNK;
    for (int i = 0; i < CHUNK; ++i) {
        const size_t l   = l0 + i;
        const float  dtv = dt[l * DMOD + d];
        const float  xv  = X[l * DMOD + d];
        const float* bmr = Bm + l * NST;
        #pragma unroll
        for (int n = 0; n < NST; ++n) {
            const float tmp = An[n] * dtv;
            const float a   = __expf(tmp);
            const float bb  = (a - 1.f) * (dtv * bmr[n]) / (tmp + 1e-6f) * xv;
            S[n] = fmaf(a, S[n], bb);
            P[n] *= a;
        }
    }
    const size_t idx = ((size_t)(b * NCH + c) * DMOD + d) * NST;
    #pragma unroll
    for (int n = 0; n < NST; ++n) { Pc[idx + n] = P[n]; Sc[idx + n] = S[n]; }
}

// ---------------------------------------------------------------------------
// Kernel 3 (pass B): sequential carry over the 16 chunks per (b,d,n).
// H0[c] = state entering chunk c.
// ---------------------------------------------------------------------------
__global__ __launch_bounds__(256) void s6_scan_carry(
    const float* __restrict__ Pc, const float* __restrict__ Sc,
    float* __restrict__ H0)
{
    const int tid = blockIdx.x * blockDim.x + threadIdx.x;   // BSZ*DMOD*NST
    const int n = tid % NST;
    const int d = (tid / NST) % DMOD;
    const int b = tid / (NST * DMOD);

    float h = 0.f;
    for (int c = 0; c < NCH; ++c) {
        const size_t idx = ((size_t)(b * NCH + c) * DMOD + d) * NST + n;
        H0[idx] = h;
        h = fmaf(Pc[idx], h, Sc[idx]);
    }
}

// ---------------------------------------------------------------------------
// Kernel 4 (pass C): replay each chunk with carried-in state, fuse the
// C-reduction and D*x skip, write the final output.
// ---------------------------------------------------------------------------
__global__ __launch_bounds__(256) void s6_scan_final(
    const float* __restrict__ X,  const float* __restrict__ Ap,
    const float* __restrict__ Bm, const float* __restrict__ Cmw,
    const float* __restrict__ dt, const float* __restrict__ H0,
    const float* __restrict__ Dp, float* __restrict__ out)
{
    const int tid = blockIdx.x * blockDim.x + threadIdx.x;   // BSZ*NCH*DMOD
    const int d = tid % DMOD;
    const int c = (tid / DMOD) % NCH;
    const int b = tid / (DMOD * NCH);

    float An[NST];
    #pragma unroll
    for (int n = 0; n < NST; ++n) An[n] = Ap[d * NST + n];

    float h[NST];
    const size_t cidx = ((size_t)(b * NCH + c) * DMOD + d) * NST;
    #pragma unroll
    for (int n = 0; n < NST; ++n) h[n] = H0[cidx + n];

    const float  Dd = Dp[d];
    const size_t l0 = (size_t)b * LSEQ + (size_t)c * CHUNK;
    for (int i = 0; i < CHUNK; ++i) {
        const size_t l   = l0 + i;
        const float  dtv = dt[l * DMOD + d];
        const float  xv  = X[l * DMOD + d];
        const float* bmr = Bm  + l * NST;
        const float* cmr = Cmw + l * NST;
        float y = 0.f;
        #pragma unroll
        for (int n = 0; n < NST; ++n) {
            const float tmp = An[n] * dtv;
            const float a   = __expf(tmp);
            const float bb  = (a - 1.f) * (dtv * bmr[n]) / (tmp + 1e-6f) * xv;
            h[n] = fmaf(a, h[n], bb);
            y    = fmaf(h[n], cmr[n], y);
        }
        out[l * DMOD + d] = fmaf(Dd, xv, y);
    }
}

// ---------------------------------------------------------------------------
// Host-side launcher.
// Inputs (setup_inputs order):
//   0:x (B,L,D) 1:A (D,N) 2:W_bc (2N,D) 3:b_bc (2N) 4:W_dt1 (R,D)
//   5:b_dt1 (R) 6:W_dt2 (D,R) 7:b_dt2 (D) 8:D_param (D)
// Workspace layout (floats):
//   Bm[B*L*N] | Cm[B*L*N] | dt[B*L*D] | P[B*NCH*D*N] | S[...] | H0[...]
// ---------------------------------------------------------------------------
extern "C" void kernel_launch(void* const* d_in, const int* in_sizes, int n_in,
                              void* d_out, int out_size, void* d_ws, size_t ws_size,
                              hipStream_t stream)
{
    const float* X    = (const float*)d_in[0];
    const float* Ap   = (const float*)d_in[1];
    const float* Wbc  = (const float*)d_in[2];
    const float* bbc  = (const float*)d_in[3];
    const float* Wdt1 = (const float*)d_in[4];
    const float* bdt1 = (const float*)d_in[5];
    const float* Wdt2 = (const float*)d_in[6];
    const float* bdt2 = (const float*)d_in[7];
    const float* Dp   = (const float*)d_in[8];
    float*       out  = (float*)d_out;

    float* ws = (float*)d_ws;
    const size_t szBC  = (size_t)MROWS * NST;            // 65536
    const size_t szDT  = (size_t)MROWS * DMOD;           // 2097152
    const size_t szCAR = (size_t)BSZ * NCH * DMOD * NST; // 262144
    float* Bm = ws;
    float* Cm = Bm + szBC;
    float* dt = Cm + szBC;
    float* Pc = dt + szDT;
    float* Sc = Pc + szCAR;
    float* H0 = Sc + szCAR;

    // 1) fused projections (WMMA f32): 256 M-tiles, 4 waves/block
    s6_proj_kernel<<<64, 128, 0, stream>>>(X, Wbc, bbc, Wdt1, bdt1, Wdt2, bdt2,
                                           Bm, Cm, dt);
    // 2) chunked scan: partial per chunk
    s6_scan_partial<<<(BSZ * NCH * DMOD) / 256, 256, 0, stream>>>(X, Ap, Bm, dt, Pc, Sc);
    // 3) carry across chunks
    s6_scan_carry<<<(BSZ * DMOD * NST) / 256, 256, 0, stream>>>(Pc, Sc, H0);
    // 4) replay with carries, fuse output
    s6_scan_final<<<(BSZ * NCH * DMOD) / 256, 256, 0, stream>>>(X, Ap, Bm, Cm, dt, H0, Dp, out);
}